// OT_Loss_77592879170068
// MI455X (gfx1250) — compile-verified
//
#include <hip/hip_runtime.h>
#include <math.h>

typedef __attribute__((ext_vector_type(2))) float v2f;
typedef __attribute__((ext_vector_type(8))) float v8f;

#define ODIM   96
#define LDS_LD 97
#define NFLAT  9216
#define NTHR   384
#define REG_F  10.0f
#define EPS_F  1e-16f

// ---------- block reductions (384 = 128*3 threads) ----------
__device__ __forceinline__ float block_reduce_sum(float v, float* red, int tid) {
  red[tid] = v;
  __syncthreads();
  if (tid < 128) red[tid] += red[tid + 128] + red[tid + 256];
  __syncthreads();
  for (int s = 64; s > 0; s >>= 1) {
    if (tid < s) red[tid] += red[tid + s];
    __syncthreads();
  }
  float r = red[0];
  __syncthreads();
  return r;
}

__device__ __forceinline__ float block_reduce_min(float v, float* red, int tid) {
  red[tid] = v;
  __syncthreads();
  if (tid < 128) red[tid] = fminf(red[tid], fminf(red[tid + 128], red[tid + 256]));
  __syncthreads();
  for (int s = 64; s > 0; s >>= 1) {
    if (tid < s) red[tid] = fminf(red[tid], red[tid + s]);
    __syncthreads();
  }
  float r = red[0];
  __syncthreads();
  return r;
}

// ---------- 96x96x96 GEMM on LDS via V_WMMA_F32_16X16X4_F32 ----------
// out[m,n] = sum_k A(m,k) * B(k,n)
//   TA: A(m,k) = Am[k*LD+m]  else Am[m*LD+k]
//   TB: B(k,n) = Bm[n*LD+k]  else Bm[k*LD+n]
//   AGD: A(m,k) = (dis[m*9216 + k*96] - e00) * Am[m*LD+k]   (A = E .* G on the fly)
// EPI: 0 = store raw to Dst
//      1 = Dst[idx] = softmax_prob(nflat) / (acc + eps)   (prob recomputed on the fly)
//      2 = return partial sum of Ud[idx] * acc            (dot with U for wd)
template<int TA, int TB, int EPI, int AGD>
__device__ __forceinline__ float gemm96(
    const float* __restrict__ Am, const float* __restrict__ Bm,
    float* __restrict__ Dst,
    const float* __restrict__ epPtr, float epM, float epInv,
    const float* __restrict__ Ud,
    const float* __restrict__ disp, float e00,
    int tid)
{
  const int lane = tid & 31;
  const int wid  = tid >> 5;          // 0..11 waves
  const int half = lane >> 4;         // 0/1
  const int lm   = lane & 15;
  const int m0   = (wid >> 1) * 16;   // row-tile base (6 row tiles, 2 waves each)
  const int nb   = (wid & 1) * 3;     // 3 col tiles per wave

  v8f acc0 = {}, acc1 = {}, acc2 = {};

  for (int k0 = 0; k0 < ODIM; k0 += 4) {
    const int ka = k0 + half * 2;
    const int am = m0 + lm;
    v2f a;
    if (AGD) {
      a.x = (disp[(size_t)am * NFLAT + (size_t)(ka + 0) * ODIM] - e00) * Am[am * LDS_LD + (ka + 0)];
      a.y = (disp[(size_t)am * NFLAT + (size_t)(ka + 1) * ODIM] - e00) * Am[am * LDS_LD + (ka + 1)];
    } else if (TA) {
      a.x = Am[(ka + 0) * LDS_LD + am];
      a.y = Am[(ka + 1) * LDS_LD + am];
    } else {
      a.x = Am[am * LDS_LD + (ka + 0)];
      a.y = Am[am * LDS_LD + (ka + 1)];
    }
    v2f b0, b1, b2;
    {
      const int n = (nb + 0) * 16 + lm;
      b0.x = TB ? Bm[n * LDS_LD + (ka + 0)] : Bm[(ka + 0) * LDS_LD + n];
      b0.y = TB ? Bm[n * LDS_LD + (ka + 1)] : Bm[(ka + 1) * LDS_LD + n];
    }
    {
      const int n = (nb + 1) * 16 + lm;
      b1.x = TB ? Bm[n * LDS_LD + (ka + 0)] : Bm[(ka + 0) * LDS_LD + n];
      b1.y = TB ? Bm[n * LDS_LD + (ka + 1)] : Bm[(ka + 1) * LDS_LD + n];
    }
    {
      const int n = (nb + 2) * 16 + lm;
      b2.x = TB ? Bm[n * LDS_LD + (ka + 0)] : Bm[(ka + 0) * LDS_LD + n];
      b2.y = TB ? Bm[n * LDS_LD + (ka + 1)] : Bm[(ka + 1) * LDS_LD + n];
    }
    acc0 = __builtin_amdgcn_wmma_f32_16x16x4_f32(false, a, false, b0, (short)0, acc0, false, false);
    acc1 = __builtin_amdgcn_wmma_f32_16x16x4_f32(false, a, false, b1, (short)0, acc1, false, false);
    acc2 = __builtin_amdgcn_wmma_f32_16x16x4_f32(false, a, false, b2, (short)0, acc2, false, false);
  }

  float wdp = 0.0f;
#pragma unroll
  for (int t = 0; t < 3; ++t) {
    const int n = (nb + t) * 16 + lm;
    v8f av = (t == 0) ? acc0 : ((t == 1) ? acc1 : acc2);
#pragma unroll
    for (int r = 0; r < 8; ++r) {
      const int m   = m0 + r + 8 * half;
      const int idx = m * LDS_LD + n;
      const float x = av[r];
      if (EPI == 0) {
        Dst[idx] = x;
      } else if (EPI == 1) {
        const int nf  = m * ODIM + n;
        const float p = expf(-epPtr[nf] - epM) * epInv;
        Dst[idx] = p / (x + EPS_F);
      } else {
        wdp += Ud[idx] * x;
      }
    }
  }
  return wdp;
}

// ---------- one workgroup per batch sample ----------
__global__ void __launch_bounds__(NTHR)
ot_sinkhorn_kernel(const float* __restrict__ normed,
                   const float* __restrict__ unnormed,
                   const float* __restrict__ gtd,
                   const float* __restrict__ dis,
                   float* __restrict__ out)
{
  __shared__ float sG[ODIM * LDS_LD];
  __shared__ float sU[ODIM * LDS_LD];
  __shared__ float sV[ODIM * LDS_LD];
  __shared__ float sS[ODIM * LDS_LD];
  __shared__ float sRed[NTHR];

  const int b   = blockIdx.x;
  const int tid = threadIdx.x;
  const float* un  = unnormed + (size_t)b * NFLAT;
  const float* gtb = gtd      + (size_t)b * NFLAT;
  const float* nd  = normed   + (size_t)b * NFLAT;

  // dis[(i,j),(d1,d2)] = E[j,d1] + E[i,d2]  =>  E[t,d] = dis[t*9216 + d*96] - dis[0]/2
  const float e00 = dis[0] * 0.5f;

  for (int t = tid; t < ODIM * ODIM; t += NTHR) {
    const int r = t / ODIM, c = t % ODIM;
    const float e = dis[(size_t)r * NFLAT + (size_t)c * ODIM] - e00;
    sG[r * LDS_LD + c] = expf(e * (-1.0f / REG_F));   // K = G (x) G
    sU[r * LDS_LD + c] = 1.0f / NFLAT;                // u0 = 1/N
  }

  // softmax(-x) scalars: max(-x) = -min(x), then sum of exp
  float mn1 = 3.0e38f, mn2 = 3.0e38f;
  for (int n = tid; n < NFLAT; n += NTHR) {
    mn1 = fminf(mn1, un[n]);
    mn2 = fminf(mn2, gtb[n]);
  }
  const float m1 = -block_reduce_min(mn1, sRed, tid);   // for src (unnormed)
  const float m2 = -block_reduce_min(mn2, sRed, tid);   // for tgt (gt)
  float s1 = 0.f, s2 = 0.f;
  for (int n = tid; n < NFLAT; n += NTHR) {
    s1 += expf(-un[n]  - m1);
    s2 += expf(-gtb[n] - m2);
  }
  const float inv1 = 1.0f / block_reduce_sum(s1, sRed, tid);
  const float inv2 = 1.0f / block_reduce_sum(s2, sRed, tid);
  __syncthreads();

  // Sinkhorn: v = b/(K^T u + eps);  u = a/(K v + eps)
#pragma unroll 1
  for (int it = 0; it < 100; ++it) {
    // S[d2,j] = sum_i G[i,d2] U[i,j]         (TN)
    gemm96<1, 0, 0, 0>(sG, sU, sS, nullptr, 0.f, 0.f, nullptr, nullptr, 0.f, tid);
    __syncthreads();
    // V[d1,d2] = src / (sum_j G[j,d1] S[d2,j] + eps)   (TT, fused div)
    gemm96<1, 1, 1, 0>(sG, sS, sV, un, m1, inv1, nullptr, nullptr, 0.f, tid);
    __syncthreads();
    // S[j,d2] = sum_d1 G[j,d1] V[d1,d2]      (NN)
    gemm96<0, 0, 0, 0>(sG, sV, sS, nullptr, 0.f, 0.f, nullptr, nullptr, 0.f, tid);
    __syncthreads();
    // U[i,j] = tgt / (sum_d2 G[i,d2] S[j,d2] + eps)    (NT, fused div)
    gemm96<0, 1, 1, 0>(sG, sS, sU, gtb, m2, inv2, nullptr, nullptr, 0.f, tid);
    __syncthreads();
  }

  // ot_obj = sum nd * beta ; loss = sum un * (-src*(1+src)*beta), beta = 10*log(v+eps)
  float pot = 0.f, pls = 0.f;
  for (int n = tid; n < NFLAT; n += NTHR) {
    const int idx    = (n / ODIM) * LDS_LD + (n % ODIM);
    const float beta = REG_F * logf(sV[idx] + EPS_F);
    pot += nd[n] * beta;
    const float sp = expf(-un[n] - m1) * inv1;
    pls += un[n] * (-sp * (1.0f + sp) * beta);
  }
  const float tot = block_reduce_sum(pot, sRed, tid);
  const float tls = block_reduce_sum(pls, sRed, tid);
  if (tid == 0) {
    atomicAdd(out + 2, tot);   // ot_obj
    atomicAdd(out + 0, tls);   // loss
  }

  // wd = sum U .* (M @ v),  M@v = G * V^T * Gd^T + Gd * V^T * G^T,  Gd = E .* G
  float wdp = 0.f;
  gemm96<0, 0, 0, 1>(sG, sV, sS, nullptr, 0.f, 0.f, nullptr, dis, e00, tid);   // S = Gd*V
  __syncthreads();
  wdp += gemm96<0, 1, 2, 0>(sG, sS, nullptr, nullptr, 0.f, 0.f, sU, nullptr, 0.f, tid); // dot(U, G*S^T)
  __syncthreads();
  gemm96<0, 0, 0, 0>(sG, sV, sS, nullptr, 0.f, 0.f, nullptr, nullptr, 0.f, tid);        // S = G*V
  __syncthreads();
  wdp += gemm96<0, 1, 2, 1>(sG, sS, nullptr, nullptr, 0.f, 0.f, sU, dis, e00, tid);     // dot(U, Gd*S^T)
  const float twd = block_reduce_sum(wdp, sRed, tid);
  if (tid == 0) atomicAdd(out + 1, twd);   // wd
}

__global__ void ot_zero_kernel(float* out, int n) {
  int i = blockIdx.x * blockDim.x + threadIdx.x;
  if (i < n) out[i] = 0.0f;
}

extern "C" void kernel_launch(void* const* d_in, const int* in_sizes, int n_in,
                              void* d_out, int out_size, void* d_ws, size_t ws_size,
                              hipStream_t stream) {
  const float* normed   = (const float*)d_in[0];
  const float* unnormed = (const float*)d_in[1];
  const float* gtd      = (const float*)d_in[2];
  const float* dis      = (const float*)d_in[3];
  // d_in[4] (points, int64) is unused by the reference math.
  float* out = (float*)d_out;

  const int B = in_sizes[0] / NFLAT;   // 8
  ot_zero_kernel<<<1, 32, 0, stream>>>(out, out_size);
  ot_sinkhorn_kernel<<<B, NTHR, 0, stream>>>(normed, unnormed, gtd, dis, out);
}